// GraphConvolution_43173011259781
// MI455X (gfx1250) — compile-verified
//
#include <hip/hip_runtime.h>

#define DEG 16
#define D   128   // D_IN == D_OUT == 128

typedef __attribute__((ext_vector_type(2))) float v2f;
typedef __attribute__((ext_vector_type(4))) float v4f;
typedef __attribute__((ext_vector_type(8))) float v8f;

// Fused GCN layer: out = relu(X*W1^T + (sum_nbr X)*W2^T + b1 + b2)
// One block = 16 nodes x 128 out-channels; 8 wave32 per block,
// wave w computes the 16x16 tile for out-channels [16w, 16w+16).
__global__ __launch_bounds__(256) void gcn_fused_wmma(
    const int*   __restrict__ nbr,   // [N, DEG]
    const float* __restrict__ X,     // [N, D]
    const float* __restrict__ W1,    // [D, D]  row-major W1[o][d]
    const float* __restrict__ b1,    // [D]
    const float* __restrict__ W2,    // [D, D]
    const float* __restrict__ b2,    // [D]
    float*       __restrict__ out,   // [N, D]
    int n)
{
    __shared__ float sX[16 * D];   // this tile's own node features
    __shared__ float sA[16 * D];   // aggregated neighbor features

    const int node0 = blockIdx.x * 16;
    const int tid   = threadIdx.x;
    const int wave  = tid >> 5;    // 0..7 (wave32)
    const int lane  = tid & 31;

    // ---------------- Stage + gather-aggregate into LDS ----------------
    // Wave w handles nodes m = 2w, 2w+1. Each lane covers 16B; 32 lanes
    // cover one full 512B feature row per access.
    for (int i = 0; i < 2; ++i) {
        const int m = wave * 2 + i;
        int node = node0 + m;
        if (node >= n) node = n - 1;           // clamp (loads stay valid)

        // Async DMA the node's own row straight into LDS (no VGPR round
        // trip): GLOBAL_LOAD_ASYNC_TO_LDS_B128, tracked by ASYNCcnt.
        {
            const unsigned lds_dst =
                (unsigned)(size_t)(sX + m * D) + (unsigned)(lane * 16);
            const unsigned long long gsrc =
                (unsigned long long)(size_t)(X + (size_t)node * D + lane * 4);
            asm volatile("global_load_async_to_lds_b128 %0, %1, off"
                         :: "v"(lds_dst), "v"(gsrc)
                         : "memory");
        }

        // Gather-sum of 16 neighbor rows (needs arithmetic -> VALU path).
        v4f acc = 0.0f;
        const int* nrow = nbr + (size_t)node * DEG;
        #pragma unroll
        for (int k = 0; k < DEG; ++k) {
            const int nb = nrow[k];            // wave-uniform -> scalar load
            acc += ((const v4f*)(X + (size_t)nb * D))[lane];
        }
        ((v4f*)(sA + m * D))[lane] = acc;
    }
    // All async LDS writes from this wave must land before the barrier.
    asm volatile("s_wait_asynccnt 0x0" ::: "memory");
    __syncthreads();

    // ---------------- Dual GEMM on the tile via V_WMMA_F32_16X16X4_F32 ----
    // A (16x4 f32): lane l holds M = l&15, K-pair base 2*(l>>4) -> v2f from LDS.
    // B (4x16 f32): lane l holds N = l&15, same K-pair base     -> v2f from W.
    const int o0 = wave * 16;                  // out-channel tile base
    const int aM = lane & 15;
    const int kk = (lane >> 4) << 1;           // 0 or 2
    const int oc = o0 + (lane & 15);
    const float* w1row = W1 + (size_t)oc * D;
    const float* w2row = W2 + (size_t)oc * D;

    // Fold the (per-lane-constant) bias into the accumulator init:
    // all 8 C regs of a lane share the same output column N = lane&15.
    const float bias = b1[oc] + b2[oc];
    v8f c;
    #pragma unroll
    for (int r = 0; r < 8; ++r) c[r] = bias;

    #pragma unroll 4
    for (int k = 0; k < D; k += 4) {
        const v2f a1  = *(const v2f*)&sX[aM * D + k + kk];
        const v2f bb1 = *(const v2f*)&w1row[k + kk];
        c = __builtin_amdgcn_wmma_f32_16x16x4_f32(
                false, a1, false, bb1, (short)0, c, false, false);

        const v2f a2  = *(const v2f*)&sA[aM * D + k + kk];
        const v2f bb2 = *(const v2f*)&w2row[k + kk];
        c = __builtin_amdgcn_wmma_f32_16x16x4_f32(
                false, a2, false, bb2, (short)0, c, false, false);
    }

    // ---------------- ReLU + store ----------------
    // C/D layout: VGPR r, lane l -> M = r + 8*(l>>4), N = l&15.
    const int mhi = (lane >> 4) << 3;          // 0 or 8
    float* orow = out + (size_t)(node0 + mhi) * D + oc;
    if (node0 + 16 <= n) {
        // Full tile (the common case): straight non-temporal stores —
        // out is write-once; keep X resident in L2 instead.
        #pragma unroll
        for (int r = 0; r < 8; ++r) {
            const float v = c[r] > 0.0f ? c[r] : 0.0f;
            __builtin_nontemporal_store(v, orow + (size_t)r * D);
        }
    } else {
        #pragma unroll
        for (int r = 0; r < 8; ++r) {
            const float v = c[r] > 0.0f ? c[r] : 0.0f;
            if (node0 + mhi + r < n)
                __builtin_nontemporal_store(v, orow + (size_t)r * D);
        }
    }
}

extern "C" void kernel_launch(void* const* d_in, const int* in_sizes, int n_in,
                              void* d_out, int out_size, void* d_ws, size_t ws_size,
                              hipStream_t stream) {
    const int*   nbr = (const int*)  d_in[0];
    const float* X   = (const float*)d_in[1];
    const float* W1  = (const float*)d_in[2];
    const float* b1  = (const float*)d_in[3];
    const float* W2  = (const float*)d_in[4];
    const float* b2  = (const float*)d_in[5];
    float* out = (float*)d_out;

    const int n     = in_sizes[0] / DEG;       // N_NODES
    const int tiles = (n + 15) / 16;

    gcn_fused_wmma<<<tiles, 256, 0, stream>>>(nbr, X, W1, b1, W2, b2, out, n);
}